// SEModule_14774687498373
// MI455X (gfx1250) — compile-verified
//
#include <hip/hip_runtime.h>

typedef __attribute__((ext_vector_type(2))) float v2f;
typedef __attribute__((ext_vector_type(4))) float v4f;
typedef __attribute__((ext_vector_type(8))) float v8f;

#define C_DIM 256
#define R_DIM 64
#define HW    3136
#define HW4   784   // HW / 4

// ---------------------------------------------------------------------------
// Kernel 1: squeeze — global average pool over H*W for each (b,c) plane.
// One 256-thread block per plane; float4 (b128) loads; LDS tree reduction.
// ---------------------------------------------------------------------------
__global__ void __launch_bounds__(256) se_squeeze(const float* __restrict__ x,
                                                  float* __restrict__ s) {
    __shared__ float red[256];
    const int plane = blockIdx.x;                       // b*C + c
    const v4f* xp = (const v4f*)(x + (size_t)plane * HW);
    float sum = 0.f;
    for (int i = threadIdx.x; i < HW4; i += 256) {      // 784/256 -> 3-4 iters
        v4f v = xp[i];
        sum += (v.x + v.y) + (v.z + v.w);
    }
    red[threadIdx.x] = sum;
    __syncthreads();
    for (int off = 128; off > 0; off >>= 1) {
        if (threadIdx.x < off) red[threadIdx.x] += red[threadIdx.x + off];
        __syncthreads();
    }
    if (threadIdx.x == 0) s[plane] = red[0] * (1.0f / (float)HW);
}

// ---------------------------------------------------------------------------
// Kernel 2: excite — H = relu(S*W1^T + b1); G = sigmoid(H*W2^T + b2).
// Single 512-thread block = 16 waves, V_WMMA_F32_16X16X4_F32 per tile.
// ---------------------------------------------------------------------------
__global__ void __launch_bounds__(512) se_excite(const float* __restrict__ s,
                                                 const float* __restrict__ w1,
                                                 const float* __restrict__ b1,
                                                 const float* __restrict__ w2,
                                                 const float* __restrict__ b2,
                                                 float* __restrict__ g) {
    // +1 pad: A-tile reads of GEMM2 walk a column (stride 64) -> avoid 16-way
    // bank conflicts on the 64-bank LDS.
    __shared__ float hbuf[R_DIM][R_DIM + 1];

    const int lane = threadIdx.x & 31;
    const int wv   = threadIdx.x >> 5;      // wave id 0..15
    const int lm   = lane & 15;             // M/N index within tile
    const int kh   = (lane >> 4) << 1;      // K sub-offset: 0 (lanes<16) or 2
    const int mo   = (lane >> 4) << 3;      // C/D row offset: 0 or 8

    // ---- GEMM1: H[64x64] = S[64x256] x W1^T[256x64] -----------------------
    {
        const int m0 = (wv >> 2) << 4;      // batch-tile row
        const int n0 = (wv & 3) << 4;       // R-tile col
        v8f acc = {};
        const float* sA = s  + (size_t)(m0 + lm) * C_DIM + kh;  // A[M][K]
        const float* wB = w1 + (size_t)(n0 + lm) * C_DIM + kh;  // B[K][N]=w1[N][K]
        for (int k = 0; k < C_DIM; k += 4) {
            v2f a; a.x = sA[k]; a.y = sA[k + 1];
            v2f b; b.x = wB[k]; b.y = wB[k + 1];
            acc = __builtin_amdgcn_wmma_f32_16x16x4_f32(
                false, a, false, b, (short)0, acc, false, false);
        }
        const float bias = b1[n0 + lm];
#pragma unroll
        for (int r = 0; r < 8; ++r) {
            float h = acc[r] + bias;
            hbuf[m0 + mo + r][n0 + lm] = h > 0.f ? h : 0.f;
        }
    }
    __syncthreads();

    // ---- GEMM2: G[64x256] = H[64x64] x W2^T[64x256] -----------------------
#pragma unroll
    for (int j = 0; j < 4; ++j) {
        const int tid = (wv << 2) + j;      // 0..63 tiles (4 x 16 grid)
        const int m0  = (tid >> 4) << 4;
        const int n0  = (tid & 15) << 4;
        v8f acc = {};
        const float* wB = w2 + (size_t)(n0 + lm) * R_DIM + kh;  // B[K][N]=w2[N][K]
        for (int k = 0; k < R_DIM; k += 4) {
            v2f a; a.x = hbuf[m0 + lm][k + kh]; a.y = hbuf[m0 + lm][k + kh + 1];
            v2f b; b.x = wB[k];                 b.y = wB[k + 1];
            acc = __builtin_amdgcn_wmma_f32_16x16x4_f32(
                false, a, false, b, (short)0, acc, false, false);
        }
        const float bias = b2[n0 + lm];
#pragma unroll
        for (int r = 0; r < 8; ++r) {
            float v = acc[r] + bias;
            g[(size_t)(m0 + mo + r) * C_DIM + n0 + lm] =
                1.0f / (1.0f + __expf(-v));
        }
    }
}

// ---------------------------------------------------------------------------
// Kernel 3: scale — out = x * g[b,c]. One block per plane; scalar-broadcast
// gain; nontemporal b128 stores (output is write-once, keep L2 for x).
// ---------------------------------------------------------------------------
__global__ void __launch_bounds__(256) se_scale(const float* __restrict__ x,
                                                const float* __restrict__ g,
                                                float* __restrict__ out) {
    const int plane = blockIdx.x;
    const float gv = g[plane];                          // uniform -> scalar load
    const v4f* xp = (const v4f*)(x + (size_t)plane * HW);
    v4f* op = (v4f*)(out + (size_t)plane * HW);
    for (int i = threadIdx.x; i < HW4; i += 256) {
        v4f v = xp[i];
        v.x *= gv; v.y *= gv; v.z *= gv; v.w *= gv;
        __builtin_nontemporal_store(v, &op[i]);
    }
}

// ---------------------------------------------------------------------------
extern "C" void kernel_launch(void* const* d_in, const int* in_sizes, int n_in,
                              void* d_out, int out_size, void* d_ws, size_t ws_size,
                              hipStream_t stream) {
    const float* x  = (const float*)d_in[0];   // [64,256,56,56]
    const float* w1 = (const float*)d_in[1];   // [64,256]
    const float* b1 = (const float*)d_in[2];   // [64]
    const float* w2 = (const float*)d_in[3];   // [256,64]
    const float* b2 = (const float*)d_in[4];   // [256]
    float* out = (float*)d_out;

    float* s = (float*)d_ws;                   // [64*256] squeeze result
    float* g = s + 64 * C_DIM;                 // [64*256] gate

    const int planes = 64 * C_DIM;             // 16384
    se_squeeze<<<planes, 256, 0, stream>>>(x, s);
    se_excite<<<1, 512, 0, stream>>>(s, w1, b1, w2, b2, g);
    se_scale<<<planes, 256, 0, stream>>>(x, g, out);
}